// GGNNSum_70952859730150
// MI455X (gfx1250) — compile-verified
//
#include <hip/hip_runtime.h>

typedef float v2f __attribute__((ext_vector_type(2)));
typedef float v8f __attribute__((ext_vector_type(8)));

#define NN   40000
#define EE   320000
#define DD   128
#define ETY  4
#define GSTEPS 8
#define NCLS 46
#define NG   40
#define NPG  1000
#define DIN  100

__device__ __forceinline__ v8f wmma_f32x4(v2f a, v2f b, v8f c) {
  // V_WMMA_F32_16X16X4_F32 : D = A(16x4,f32) * B(4x16,f32) + C(16x16,f32)
  return __builtin_amdgcn_wmma_f32_16x16x4_f32(
      /*neg_a=*/false, a, /*neg_b=*/false, b,
      /*c_mod=*/(short)0, c, /*reuse_a=*/false, /*reuse_b=*/false);
}

__device__ __forceinline__ float sigmoidf_(float x) {
  return 1.0f / (1.0f + __expf(-x));
}

// ---- h0 = pad(features, [N,100] -> [N,128]) ------------------------------
__global__ void k_init_h(const float* __restrict__ feat, float* __restrict__ h) {
  int i = blockIdx.x * blockDim.x + threadIdx.x;
  if (i >= NN * DD) return;
  int n = i >> 7, d = i & 127;
  h[i] = (d < DIN) ? feat[n * DIN + d] : 0.0f;
}

// ---- Wt[et][f][d] = W_et[et][d][f]  (one-time, makes B-frags contiguous) -
__global__ void k_transpose_wet(const float* __restrict__ W_et,
                                float* __restrict__ Wt) {
  int i = blockIdx.x * blockDim.x + threadIdx.x;   // < ETY*DD*DD
  int d  = i & 127;
  int f  = (i >> 7) & 127;
  int et = i >> 14;
  Wt[i] = W_et[((size_t)et << 14) + ((size_t)d << 7) + f];
}

// ---- t[et] = h @ W_et[et] + b_et[et]   ([N,128]x[128,128], f32 WMMA) -----
// grid: (N/16, ETY), block: 256 (8 waves; wave w owns N-tile w)
__global__ void k_gemm_t(const float* __restrict__ h,
                         const float* __restrict__ Wt,   // [et][f][d]
                         const float* __restrict__ b_et,
                         float* __restrict__ t) {
  const int wave = threadIdx.x >> 5;
  const int lane = threadIdx.x & 31;
  const int hf   = lane >> 4;       // half of wave
  const int idx  = lane & 15;
  const int m0   = blockIdx.x << 4;
  const int et   = blockIdx.y;
  const int f    = (wave << 4) + idx;               // output column

  const float bias = b_et[et * DD + f];
  v8f acc;
#pragma unroll
  for (int r = 0; r < 8; ++r) acc[r] = bias;

  const float* Arow = h + (size_t)(m0 + idx) * DD;            // A row (this lane)
  const float* Brow = Wt + ((size_t)et << 14) + ((size_t)f << 7); // B col, contiguous in k
#pragma unroll 4
  for (int k = 0; k < DD; k += 4) {
    const int kk = k + (hf << 1);
    v2f va = *(const v2f*)(Arow + kk);              // A[m, kk..kk+1]
    v2f vb = *(const v2f*)(Brow + kk);              // B[kk..kk+1, f]
    acc = wmma_f32x4(va, vb, acc);
  }

  float* Tout = t + ((size_t)et * NN + m0) * DD + f;
#pragma unroll
  for (int r = 0; r < 8; ++r)
    Tout[(size_t)(r + (hf << 3)) * DD] = acc[r];    // row m0 + r + 8*half
}

// ---- a[dst[e]] += t[etype[e], src[e]]  (128 threads per edge) ------------
__global__ void k_scatter(const float* __restrict__ t,
                          const int* __restrict__ src,
                          const int* __restrict__ dst,
                          const int* __restrict__ ety,
                          float* __restrict__ a) {
  size_t g = (size_t)blockIdx.x * blockDim.x + threadIdx.x;
  int e = (int)(g >> 7);
  int f = (int)(g & 127);
  if (e >= EE) return;
  float v = t[((size_t)ety[e] * NN + src[e]) * DD + f];
  unsafeAtomicAdd(a + (size_t)dst[e] * DD + f, v);  // -> global_atomic_add_f32
}

// ---- fused GRU: gi = a@W_ih^T+b_ih, gh = h@W_hh^T+b_hh, gate, write hn ---
// grid: N/16, block: 256 (wave j owns output cols [16j,16j+16) of each gate)
__global__ void k_gru(const float* __restrict__ a,
                      const float* __restrict__ h,
                      const float* __restrict__ W_ih,
                      const float* __restrict__ b_ih,
                      const float* __restrict__ W_hh,
                      const float* __restrict__ b_hh,
                      float* __restrict__ hn) {
  const int j    = threadIdx.x >> 5;
  const int lane = threadIdx.x & 31;
  const int hf   = lane >> 4;
  const int idx  = lane & 15;
  const int m0   = blockIdx.x << 4;
  const int col  = (j << 4) + idx;                  // 0..127 within each gate

  v8f cir, ciz, cin, chr_, chz, chn;
  {
    const float bir = b_ih[col], biz = b_ih[DD + col], bin_ = b_ih[2 * DD + col];
    const float bhr = b_hh[col], bhz = b_hh[DD + col], bhn  = b_hh[2 * DD + col];
#pragma unroll
    for (int r = 0; r < 8; ++r) {
      cir[r] = bir;  ciz[r] = biz;  cin[r] = bin_;
      chr_[r] = bhr; chz[r] = bhz;  chn[r] = bhn;
    }
  }

  const float* Aa  = a + (size_t)(m0 + idx) * DD;
  const float* Ah  = h + (size_t)(m0 + idx) * DD;
  // gi[n,o] = sum_k a[n,k]*W_ih[o,k]  ->  B[k,o] = W_ih[o,k] (contiguous in k)
  const float* Wir = W_ih + (size_t)col * DD;
  const float* Wiz = W_ih + (size_t)(DD + col) * DD;
  const float* Win = W_ih + (size_t)(2 * DD + col) * DD;
  const float* Whr = W_hh + (size_t)col * DD;
  const float* Whz = W_hh + (size_t)(DD + col) * DD;
  const float* Whn = W_hh + (size_t)(2 * DD + col) * DD;

#pragma unroll 2
  for (int k = 0; k < DD; k += 4) {
    const int kk = k + (hf << 1);
    v2f va = *(const v2f*)(Aa + kk);
    v2f vh = *(const v2f*)(Ah + kk);
    v2f vb;
    vb = *(const v2f*)(Wir + kk); cir  = wmma_f32x4(va, vb, cir);
    vb = *(const v2f*)(Wiz + kk); ciz  = wmma_f32x4(va, vb, ciz);
    vb = *(const v2f*)(Win + kk); cin  = wmma_f32x4(va, vb, cin);
    vb = *(const v2f*)(Whr + kk); chr_ = wmma_f32x4(vh, vb, chr_);
    vb = *(const v2f*)(Whz + kk); chz  = wmma_f32x4(vh, vb, chz);
    vb = *(const v2f*)(Whn + kk); chn  = wmma_f32x4(vh, vb, chn);
  }

#pragma unroll
  for (int r = 0; r < 8; ++r) {
    const int m    = m0 + r + (hf << 3);
    const float rg  = sigmoidf_(cir[r] + chr_[r]);
    const float z   = sigmoidf_(ciz[r] + chz[r]);
    const float nn_ = tanhf(cin[r] + rg * chn[r]);
    const float ho  = h[(size_t)m * DD + col];
    hn[(size_t)m * DD + col] = (1.0f - z) * nn_ + z * ho;
  }
}

// ---- h_i = tanh(h) into output region ------------------------------------
__global__ void k_tanh_out(const float* __restrict__ h, float* __restrict__ o) {
  int i = blockIdx.x * blockDim.x + threadIdx.x;
  if (i < NN * DD) o[i] = tanhf(h[i]);
}

// ---- per-graph feature sum of h_i ----------------------------------------
__global__ void k_gsum(const float* __restrict__ hi, float* __restrict__ s) {
  const int g = blockIdx.x;
  const int d = threadIdx.x;
  float acc = 0.0f;
  const float* p = hi + (size_t)g * NPG * DD + d;
  for (int i = 0; i < NPG; ++i) acc += p[(size_t)i * DD];
  s[g * DD + d] = acc;
}

// ---- logits[g,c] = s[g] . W_cls[c] + b_cls[c] ----------------------------
__global__ void k_logits(const float* __restrict__ s,
                         const float* __restrict__ Wc,
                         const float* __restrict__ bc,
                         float* __restrict__ out) {
  const int g = blockIdx.x;
  const int c = threadIdx.x;
  if (c >= NCLS) return;
  float acc = bc[c];
  for (int d = 0; d < DD; ++d) acc += s[g * DD + d] * Wc[c * DD + d];
  out[g * NCLS + c] = acc;
}

extern "C" void kernel_launch(void* const* d_in, const int* in_sizes, int n_in,
                              void* d_out, int out_size, void* d_ws, size_t ws_size,
                              hipStream_t stream) {
  const float* feat  = (const float*)d_in[0];
  const float* W_et  = (const float*)d_in[1];
  const float* b_et  = (const float*)d_in[2];
  const float* W_ih  = (const float*)d_in[3];
  const float* b_ih  = (const float*)d_in[4];
  const float* W_hh  = (const float*)d_in[5];
  const float* b_hh  = (const float*)d_in[6];
  const float* W_cls = (const float*)d_in[7];
  const float* b_cls = (const float*)d_in[8];
  const int*   src   = (const int*)d_in[9];
  const int*   dst   = (const int*)d_in[10];
  const int*   ety   = (const int*)d_in[11];

  char* ws = (char*)d_ws;
  const size_t HS = (size_t)NN * DD * sizeof(float);        // 20.48 MB
  float* h0 = (float*)(ws);
  float* h1 = (float*)(ws + HS);
  float* t  = (float*)(ws + 2 * HS);                        // ETY * HS
  float* a  = (float*)(ws + (2 + ETY) * HS);
  float* s  = (float*)(ws + (3 + ETY) * HS);                // NG*DD floats
  float* Wt = (float*)(ws + (3 + ETY) * HS + (size_t)NG * DD * sizeof(float));

  float* logits = (float*)d_out;                            // [40,46]
  float* hi     = logits + (size_t)NG * NCLS;               // [40,1000,128]

  k_transpose_wet<<<(ETY * DD * DD) / 256, 256, 0, stream>>>(W_et, Wt);
  k_init_h<<<(NN * DD + 255) / 256, 256, 0, stream>>>(feat, h0);

  float* hc = h0;
  float* hx = h1;
  for (int step = 0; step < GSTEPS; ++step) {
    dim3 gT(NN / 16, ETY);
    k_gemm_t<<<gT, 256, 0, stream>>>(hc, Wt, b_et, t);
    hipMemsetAsync(a, 0, HS, stream);
    k_scatter<<<(unsigned)((size_t)EE * DD / 256), 256, 0, stream>>>(t, src, dst, ety, a);
    k_gru<<<NN / 16, 256, 0, stream>>>(a, hc, W_ih, b_ih, W_hh, b_hh, hx);
    float* tmp = hc; hc = hx; hx = tmp;
  }

  k_tanh_out<<<(NN * DD + 255) / 256, 256, 0, stream>>>(hc, hi);
  k_gsum<<<NG, DD, 0, stream>>>(hi, s);
  k_logits<<<NG, 64, 0, stream>>>(s, W_cls, b_cls, logits);
}